// AnalyticalAR_33681133535968
// MI455X (gfx1250) — compile-verified
//
#include <hip/hip_runtime.h>

typedef __attribute__((ext_vector_type(2))) float v2f;
typedef __attribute__((ext_vector_type(4))) float f4;
typedef __attribute__((ext_vector_type(8))) float v8f;

#define SEQ   600
#define TOUT  594   // SEQ - P
#define PP    6
#define NCLS  5
#define WPB   8     // waves per block (256 threads, wave32)

__global__ __launch_bounds__(256) void
ar_fit_kernel(const float* __restrict__ x, float* __restrict__ out, int N) {
    __shared__ float xs_all[WPB][SEQ + 8];
    __shared__ float g_all[WPB][64];

    const int lane = threadIdx.x & 31;
    const int wave = threadIdx.x >> 5;
    int row = blockIdx.x * WPB + wave;
    if (row >= N) row = N - 1;   // duplicate (identical) work; keeps barriers uniform

    float* xs = xs_all[wave];
    float* G  = g_all[wave];

    // ---- stage x row into LDS (coalesced B128 loads); zero the pad tail ----
    const f4* xrow4 = (const f4*)(x + (size_t)row * SEQ);
    f4* xs4 = (f4*)xs;
    for (int i = lane; i < SEQ / 4; i += 32) xs4[i] = xrow4[i];
    if (lane < 8) xs[SEQ + lane] = 0.0f;
    __syncthreads();

    // ---- G~ = D~^T D~ via chained V_WMMA_F32_16X16X4_F32 ----
    // Augmented design: col 0 = ones, cols 1..6 = lags x[t+P-m], col 7 = y = x[t+P].
    // A(16x4) and B(4x16) have identical per-lane register contents for B = A^T,
    // so one v2f feeds both operands. Only C rows/cols 0..7 are read back, which
    // depend only on A rows 0..7 / B cols 0..7 -> lanes m>7 may carry junk.
    const int m  = lane & 15;
    const int kb = (lane >> 4) << 1;          // K pair: lanes 0-15 -> {0,1}, 16-31 -> {2,3}
    int off;
    if (m == 0)      off = 0;                 // value replaced by 1.0 via fma
    else if (m <= 6) off = PP - m;            // lag columns
    else if (m == 7) off = PP;                // y column
    else             off = 0;                 // m 8..15: result unused
    const float sel_s = (m == 0) ? 0.0f : 1.0f;
    const float sel_b = (m == 0) ? 1.0f : 0.0f;
    const float* p = xs + off + kb;

    v8f c = {};
    int t0 = 0;
    for (; t0 < TOUT - 4; t0 += 4) {          // 148 full iterations, t <= 591 < TOUT
        float ax = fmaf(p[t0],     sel_s, sel_b);
        float ay = fmaf(p[t0 + 1], sel_s, sel_b);
        v2f a = {ax, ay};
        c = __builtin_amdgcn_wmma_f32_16x16x4_f32(
                false, a, false, a, (short)0, c, false, false);
    }
    {   // tail t0 = 592: K slots t = 594,595 (lanes 16-31) are padding -> zero
        float ax = fmaf(p[t0],     sel_s, sel_b);
        float ay = fmaf(p[t0 + 1], sel_s, sel_b);
        if (kb != 0) { ax = 0.0f; ay = 0.0f; }
        v2f a = {ax, ay};
        c = __builtin_amdgcn_wmma_f32_16x16x4_f32(
                false, a, false, a, (short)0, c, false, false);
    }

    // C layout: lanes 0-15, VGPR j -> G~[j][lane]. Keep the 8x8 block.
    if (lane < 8) {
#pragma unroll
        for (int j = 0; j < 8; ++j) G[j * 8 + lane] = c[j];
    }
    __syncthreads();

    // ---- solve G w = b : Gauss-Jordan on augmented 7x8 (col 7 = b) ----
    if (lane == 0) {
#pragma unroll 1
        for (int pv = 0; pv < 7; ++pv) {
            float inv = 1.0f / G[pv * 8 + pv];
#pragma unroll 1
            for (int j = pv; j < 8; ++j) G[pv * 8 + j] *= inv;
#pragma unroll 1
            for (int r = 0; r < 7; ++r) {
                if (r == pv) continue;
                float f = G[r * 8 + pv];
#pragma unroll 1
                for (int j = pv; j < 8; ++j) G[r * 8 + j] -= f * G[pv * 8 + j];
            }
        }
    }
    __syncthreads();

    const float w0 = G[0 * 8 + 7], w1 = G[1 * 8 + 7], w2 = G[2 * 8 + 7],
                w3 = G[3 * 8 + 7], w4 = G[4 * 8 + 7], w5 = G[5 * 8 + 7],
                w6 = G[6 * 8 + 7];

    // ---- coeffs: (row, 600, 6); zeros for t<6, else w[1..6]; 12-float period ----
    {
        f4* oc = (f4*)(out + (size_t)row * (SEQ * PP));
        f4 pat0 = {w1, w2, w3, w4};
        f4 pat1 = {w5, w6, w1, w2};
        f4 pat2 = {w3, w4, w5, w6};
        f4 zero = {};
        for (int q = lane; q < (SEQ * PP) / 4; q += 32) {   // 900 float4s
            f4 v;
            if (q < 9) v = zero;                            // t < 6 region (36 floats)
            else {
                int r = q % 3;
                v = (r == 0) ? pat0 : ((r == 1) ? pat1 : pat2);
            }
            oc[q] = v;
        }
    }

    // ---- p_logits (zeros) and p_hard (zeros; int 0 == float 0 bitwise) ----
    {
        float* ol = out + (size_t)N * (SEQ * PP) + (size_t)row * NCLS;
        if (lane < NCLS) ol[lane] = 0.0f;
        if (lane == 0)
            out[(size_t)N * (SEQ * PP + NCLS) + row] = 0.0f;
    }

    // ---- x_hat: zeros for t<6, else w0 + sum_m w[m]*x[t-m] ----
    {
        f4* oxh = (f4*)(out + (size_t)N * (SEQ * PP + NCLS + 1) + (size_t)row * SEQ);
        for (int q = lane; q < SEQ / 4; q += 32) {          // 150 float4s
            f4 v;
#pragma unroll
            for (int e = 0; e < 4; ++e) {
                int t = 4 * q + e;
                float val = 0.0f;
                if (t >= PP) {
                    val = w0 + w1 * xs[t - 1] + w2 * xs[t - 2] + w3 * xs[t - 3]
                             + w4 * xs[t - 4] + w5 * xs[t - 5] + w6 * xs[t - 6];
                }
                v[e] = val;
            }
            oxh[q] = v;
        }
    }
}

extern "C" void kernel_launch(void* const* d_in, const int* in_sizes, int n_in,
                              void* d_out, int out_size, void* d_ws, size_t ws_size,
                              hipStream_t stream) {
    const float* x = (const float*)d_in[0];
    float* out = (float*)d_out;
    const int N = in_sizes[0] / SEQ;                 // 32768
    const int blocks = (N + WPB - 1) / WPB;          // 4096 blocks of 8 waves
    ar_fit_kernel<<<blocks, 256, 0, stream>>>(x, out, N);
}